// TemporalDeepBasisChebnetLocalFilterOnGraphKernel_24077586661964
// MI455X (gfx1250) — compile-verified
//
#include <hip/hip_runtime.h>

typedef float v2f __attribute__((ext_vector_type(2)));
typedef float v8f __attribute__((ext_vector_type(8)));

#define NBT 4
#define WID 5
#define NNODE 4096

// softplus(100*z)/100, numerically stable
__device__ __forceinline__ float sp100(float z) {
    float u = 100.0f * z;
    float e = __expf(-fabsf(u));
    return (fmaxf(u, 0.0f) + __logf(1.0f + e)) * 0.01f;
}

// 4 stacked MLPs: 1 -> 5 -> 5 -> 1, params indexed by basis n
__device__ __forceinline__ void mlp4(float t,
                                     const float* __restrict__ W1,
                                     const float* __restrict__ b1,
                                     const float* __restrict__ W2,
                                     const float* __restrict__ b2,
                                     const float* __restrict__ W3,
                                     const float* __restrict__ b3,
                                     float o[NBT]) {
#pragma unroll
    for (int n = 0; n < NBT; ++n) {
        float h1[WID], h2[WID];
#pragma unroll
        for (int w = 0; w < WID; ++w)
            h1[w] = sp100(fmaf(t, W1[n * WID + w], b1[n * WID + w]));
#pragma unroll
        for (int v = 0; v < WID; ++v) {
            float acc = b2[n * WID + v];
#pragma unroll
            for (int w = 0; w < WID; ++w)
                acc = fmaf(h1[w], W2[n * WID * WID + w * WID + v], acc);
            h2[v] = sp100(acc);
        }
        float acc = b3[n];
#pragma unroll
        for (int w = 0; w < WID; ++w)
            acc = fmaf(h2[w], W3[n * WID + w], acc);
        o[n] = acc;
    }
}

__global__ __launch_bounds__(256) void tdb_cheb_kernel(
    const float* __restrict__ x, const float* __restrict__ y,
    const float* __restrict__ Bm,
    const float* __restrict__ Wx1, const float* __restrict__ Wx2, const float* __restrict__ Wx3,
    const float* __restrict__ Wy1, const float* __restrict__ Wy2, const float* __restrict__ Wy3,
    const float* __restrict__ bx1, const float* __restrict__ bx2, const float* __restrict__ bx3,
    const float* __restrict__ by1, const float* __restrict__ by2, const float* __restrict__ by3,
    const float* __restrict__ Wij,
    float* __restrict__ out, int batch)
{
    __shared__ float Wl[16];
    __shared__ __align__(16) float kts[256][4];  // kt per sample (WMMA A staging)
    __shared__ __align__(16) float qls[256][4];  // q = kt @ W per sample (WMMA D landing)

    const int tid = threadIdx.x;
    const int b   = blockIdx.x * 256 + tid;
    const bool inb = (b < batch);

    if (tid < 16) Wl[tid] = Wij[tid];

    // ---- load sample, issue gathers early (overlap with MLP compute) ----
    float tx = 0.0f, ty = 0.0f;
    int   xi = 0,    yi = 0;
    if (inb) {
        float2 xv = ((const float2*)x)[b];
        float2 yv = ((const float2*)y)[b];
        tx = xv.x; xi = (int)xv.y;
        ty = yv.x; yi = (int)yv.y;
    }
    const float dt   = tx - ty;
    const bool  live = inb && (dt <= 10.0f);

    float kl0 = 0.0f, kl1 = 0.0f, kl2 = 0.0f, kl3 = 0.0f;
    float kt0 = 0.0f, kt1 = 0.0f, kt2 = 0.0f, kt3 = 0.0f;
    if (live) {
        const size_t base  = (size_t)yi * NNODE + (size_t)xi;
        const size_t slice = (size_t)NNODE * NNODE;
        kl0 = Bm[base];              // gathers in flight while MLPs run
        kl1 = Bm[base + slice];
        kl2 = Bm[base + 2 * slice];
        kl3 = Bm[base + 3 * slice];

        float xb[NBT], yb[NBT];
        mlp4(dt * 0.01f, Wx1, bx1, Wx2, bx2, Wx3, bx3, xb);  // (x-t - y-t)/(T1-T0)
        mlp4(ty * 0.01f, Wy1, by1, Wy2, by2, Wy3, by3, yb);  // (y-t - T0)/(T1-T0)
        kt0 = xb[0] * yb[0];
        kt1 = xb[1] * yb[1];
        kt2 = xb[2] * yb[2];
        kt3 = xb[3] * yb[3];
    }

    // ---- stage kt into LDS for WMMA A-matrix marshaling ----
    *(float4*)kts[tid] = make_float4(kt0, kt1, kt2, kt3);
    __syncthreads();

    const int  lane  = tid & 31;
    const int  wbase = tid & ~31;       // first sample of this wave
    const int  ncol  = lane & 15;       // WMMA N index
    const bool lo    = lane < 16;

    // B-matrix (4x16): B[i][n] = W[i][n] for n<4 else 0.
    // 32-bit layout: VGPR0 = rows {0,2} (lane halves), VGPR1 = rows {1,3}.
    v2f bmat;
    bmat.x = (ncol < 4) ? Wl[(lo ? 0 : 2) * 4 + ncol] : 0.0f;
    bmat.y = (ncol < 4) ? Wl[(lo ? 1 : 3) * 4 + ncol] : 0.0f;

    // Two 16x16x4 f32 WMMAs per wave: D[m, j] = sum_i kt[m, i] * W[i, j]
#pragma unroll
    for (int g = 0; g < 2; ++g) {
        const int src = wbase + 16 * g + ncol;  // A row m = ncol of group g
        v2f a;
        a.x = kts[src][lo ? 0 : 2];   // A VGPR0: K=0 (lanes 0-15) / K=2 (lanes 16-31)
        a.y = kts[src][lo ? 1 : 3];   // A VGPR1: K=1 / K=3
        v8f c = {0.f, 0.f, 0.f, 0.f, 0.f, 0.f, 0.f, 0.f};
        v8f d = __builtin_amdgcn_wmma_f32_16x16x4_f32(
            /*neg_a=*/false, a, /*neg_b=*/false, bmat,
            /*c_mod=*/(short)0, c, /*reuse_a=*/false, /*reuse_b=*/false);
        if (ncol < 4) {
            // D layout: VGPR r -> M = r (lanes 0-15) / M = r+8 (lanes 16-31), N = ncol
#pragma unroll
            for (int r = 0; r < 8; ++r) {
                const int M = r + (lo ? 0 : 8);
                qls[wbase + 16 * g + M][ncol] = d[r];
            }
        }
    }
    __syncthreads();

    if (inb) {
        float q0 = qls[tid][0], q1 = qls[tid][1], q2 = qls[tid][2], q3 = qls[tid][3];
        out[b] = fmaf(q0, kl0, fmaf(q1, kl1, fmaf(q2, kl2, q3 * kl3)));
    }
}

extern "C" void kernel_launch(void* const* d_in, const int* in_sizes, int n_in,
                              void* d_out, int out_size, void* d_ws, size_t ws_size,
                              hipStream_t stream) {
    const float* x   = (const float*)d_in[0];
    const float* y   = (const float*)d_in[1];
    const float* Bm  = (const float*)d_in[2];
    const float* Wx1 = (const float*)d_in[3];
    const float* Wx2 = (const float*)d_in[4];
    const float* Wx3 = (const float*)d_in[5];
    const float* Wy1 = (const float*)d_in[6];
    const float* Wy2 = (const float*)d_in[7];
    const float* Wy3 = (const float*)d_in[8];
    const float* bx1 = (const float*)d_in[9];
    const float* bx2 = (const float*)d_in[10];
    const float* bx3 = (const float*)d_in[11];
    const float* by1 = (const float*)d_in[12];
    const float* by2 = (const float*)d_in[13];
    const float* by3 = (const float*)d_in[14];
    const float* Wij = (const float*)d_in[15];
    float* out = (float*)d_out;

    const int batch  = in_sizes[0] / 2;          // x is [B, 2]
    const int blocks = (batch + 255) / 256;      // 1048576/256 = 4096 exact

    hipLaunchKernelGGL(tdb_cheb_kernel, dim3(blocks), dim3(256), 0, stream,
                       x, y, Bm, Wx1, Wx2, Wx3, Wy1, Wy2, Wy3,
                       bx1, bx2, bx3, by1, by2, by3, Wij, out, batch);
}